// TemporalDecay_89524298318172
// MI455X (gfx1250) — compile-verified
//
#include <hip/hip_runtime.h>

// Problem constants from the reference: B=32, T=2048, D=64, K=4 (K*D = 256)
#define TT      2048
#define DD      64
#define KDIM    256          // K*D floats per time row
#define TILE_T  32           // time rows computed per workgroup
#define HALO    3            // delta in [1,4] -> gather reaches back at most 3 rows
#define SROWS   (TILE_T + HALO)          // 35 staged rows
#define STAGE_CHUNKS (SROWS * (KDIM/4)) // 2240 float4 chunks staged per block
#define COMP_CHUNKS  (TILE_T * (KDIM/4))// 2048 float4 chunks computed per block

typedef __attribute__((address_space(3))) char lds_char_t;

// h[b,t,kd] = h_a + (1-m)*gamma*(h_fwd - h_a),  gamma = exp(-relu(delta*W+b)),
// h_fwd[b,t,kd] = h_a[b, t-(delta-1), kd].
// h_a tile (TILE_T rows + 3 halo rows) is staged into LDS with the CDNA5 async
// copy engine (global_load_async_to_lds_b128 / ASYNCcnt); both the streaming
// read and the time-gather are then served from LDS.
__global__ __launch_bounds__(256) void TemporalDecay_kernel(
    const float* __restrict__ h_a,
    const float* __restrict__ deltas,
    const float* __restrict__ M,
    const float* __restrict__ W,
    const float* __restrict__ bias,
    float* __restrict__ out)
{
    __shared__ __align__(16) float sm[SROWS * KDIM];   // 35 KB

    const int blk = blockIdx.x;            // b * (T/TILE_T) + tile
    const int b   = blk >> 6;              // T/TILE_T = 64
    const int t0  = (blk & 63) * TILE_T;
    const long bT = (long)b * TT;

    // LDS byte offset of sm[0] (addrspace(3) pointers are 32-bit LDS offsets)
    const unsigned lds_base =
        (unsigned)(unsigned long long)(lds_char_t*)sm;

    // ---- Stage h_a rows [t0-3, t0+TILE_T) into LDS via async copy engine ----
    for (int i = threadIdx.x; i < STAGE_CHUNKS; i += 256) {
        const int s = i >> 6;              // staged slot 0..34
        const int c = i & 63;              // float4 chunk within row
        const int r = t0 - HALO + s;       // global time row
        if (r >= 0) {                      // slots 0..2 unused when t0==0 (never read)
            const float* src = h_a + (((size_t)(bT + r)) << 8) + (c << 2);
            const unsigned lds_off = lds_base + ((unsigned)i << 4);  // i * 16 bytes
            asm volatile("global_load_async_to_lds_b128 %0, %1, off"
                         :: "v"(lds_off), "v"(src)
                         : "memory");
        }
    }
    asm volatile("s_wait_asynccnt 0x0" ::: "memory");
    __syncthreads();

    // ---- Compute: 8 float4 chunks per thread (2048 chunks / 256 threads) ----
    for (int i = threadIdx.x; i < COMP_CHUNKS; i += 256) {
        const int rt  = i >> 6;            // row within tile
        const int c   = i & 63;
        const int kd0 = c << 2;            // [0,256), multiple of 4
        const int d0  = kd0 & (DD - 1);    // delta/mask feature index = kd % D
        const size_t row = (size_t)(bT + t0 + rt);

        const float4 ha = *reinterpret_cast<const float4*>(&sm[(rt + HALO) * KDIM + kd0]);
        const float4 dl = *reinterpret_cast<const float4*>(deltas + row * DD + d0);
        const float4 mm = *reinterpret_cast<const float4*>(M      + row * DD + d0);
        const float4 w4 = *reinterpret_cast<const float4*>(W    + kd0);
        const float4 b4 = *reinterpret_cast<const float4*>(bias + kd0);

        // LDS slot of gather row: (t-(delta-1)) - (t0-HALO) = rt + 4 - delta
        const int s0 = rt + 4 - __float2int_rn(dl.x);
        const int s1 = rt + 4 - __float2int_rn(dl.y);
        const int s2 = rt + 4 - __float2int_rn(dl.z);
        const int s3 = rt + 4 - __float2int_rn(dl.w);

        const float hf0 = sm[s0 * KDIM + kd0 + 0];   // conflict-free: dword addr = 4*lane+j
        const float hf1 = sm[s1 * KDIM + kd0 + 1];
        const float hf2 = sm[s2 * KDIM + kd0 + 2];
        const float hf3 = sm[s3 * KDIM + kd0 + 3];

        const float g0 = __expf(-fmaxf(fmaf(dl.x, w4.x, b4.x), 0.0f));
        const float g1 = __expf(-fmaxf(fmaf(dl.y, w4.y, b4.y), 0.0f));
        const float g2 = __expf(-fmaxf(fmaf(dl.z, w4.z, b4.z), 0.0f));
        const float g3 = __expf(-fmaxf(fmaf(dl.w, w4.w, b4.w), 0.0f));

        // h = m*ha + (1-m)*(g*hf + (1-g)*ha) == ha + (1-m)*g*(hf - ha)
        float4 o;
        o.x = fmaf((1.0f - mm.x) * g0, hf0 - ha.x, ha.x);
        o.y = fmaf((1.0f - mm.y) * g1, hf1 - ha.y, ha.y);
        o.z = fmaf((1.0f - mm.z) * g2, hf2 - ha.z, ha.z);
        o.w = fmaf((1.0f - mm.w) * g3, hf3 - ha.w, ha.w);

        *reinterpret_cast<float4*>(out + row * KDIM + kd0) = o;
    }
}

extern "C" void kernel_launch(void* const* d_in, const int* in_sizes, int n_in,
                              void* d_out, int out_size, void* d_ws, size_t ws_size,
                              hipStream_t stream) {
    const float* h_a    = (const float*)d_in[0];  // [B, T, K*D]
    const float* deltas = (const float*)d_in[1];  // [B, T, D]
    const float* M      = (const float*)d_in[2];  // [B, T, D]
    const float* W      = (const float*)d_in[3];  // [K*D]
    const float* bias   = (const float*)d_in[4];  // [K*D]
    float* out = (float*)d_out;                   // [B, T, K*D]

    const int blocks = 32 * (TT / TILE_T);        // B * 64 = 2048 workgroups
    TemporalDecay_kernel<<<blocks, 256, 0, stream>>>(h_a, deltas, M, W, bias, out);
}